// PNALocal_5214090297741
// MI455X (gfx1250) — compile-verified
//
#include <hip/hip_runtime.h>
#include <math.h>

#define N_NODES 20000
#define N_EDGES 320000
#define HDIM    64
#define TDIM    256
#define NLAYERS 5
#define KPRE    192            // 2H + EH
#define KPOST   832            // 13H
#define AVG_D_LOG 2.8332133440562162f
#define EPSV    1e-5f
#define TPB_TILES 8            // edge tiles per persistent pretrans block

typedef __attribute__((ext_vector_type(2))) float v2f;
typedef __attribute__((ext_vector_type(8))) float v8f;

// ---- wave-level 16x16 fp32 WMMA GEMM, packed-pair B ----
// A: LDS tile [16 x K], row stride lda. Bp: packed pairs, entry (s,hh,n) at
// (s*2+hh)*ldb + n holds {B[4s+2hh][n], B[4s+2hh+1][n]} -> one b64 load/step.
__device__ __forceinline__ v8f wmma_gemm_pk(const float* A, int lda,
                                            const float2* __restrict__ Bp, int ldb,
                                            int nOff, int K) {
  const int lane = threadIdx.x & 31;
  const int m  = lane & 15;
  const int kh = (lane >> 4) << 1;            // 0 or 2
  const int hh = lane >> 4;                   // 0 or 1
  const int n  = nOff + (lane & 15);
  v8f c = {0.f, 0.f, 0.f, 0.f, 0.f, 0.f, 0.f, 0.f};
#pragma unroll 4
  for (int k0 = 0; k0 < K; k0 += 4) {
    const int s = k0 >> 2;
    v2f a, b;
    a.x = A[m * lda + k0 + kh];
    a.y = A[m * lda + k0 + kh + 1];
    const float2 w = Bp[(size_t)(s * 2 + hh) * ldb + n];
    b.x = w.x; b.y = w.y;
    c = __builtin_amdgcn_wmma_f32_16x16x4_f32(false, a, false, b, (short)0, c,
                                              false, false);
  }
  return c;
}

// repack B[K][ldb] (row-major) into fragment-pair order
__global__ void k_pack(const float* __restrict__ B, float2* __restrict__ Bp,
                       int K, int ldb) {
  const int total = (K >> 2) * 2 * ldb;
  int i = blockIdx.x * 256 + threadIdx.x;
  if (i < total) {
    const int n  = i % ldb;
    const int sh = i / ldb;
    const int k  = 4 * (sh >> 1) + 2 * (sh & 1);
    float2 v; v.x = B[(size_t)k * ldb + n]; v.y = B[(size_t)(k + 1) * ldb + n];
    Bp[i] = v;
  }
}

// ================= CSR build (counting sort by dst, once per call) ============

__global__ void k_hist(const int* __restrict__ dst, unsigned* __restrict__ cnt) {
  int e = blockIdx.x * blockDim.x + threadIdx.x;
  if (e < N_EDGES) atomicAdd(&cnt[dst[e]], 1u);
}

__global__ __launch_bounds__(1024) void k_scan(const unsigned* __restrict__ cnt,
                                               unsigned* __restrict__ rowstart) {
  __shared__ unsigned buf[1024];
  __shared__ unsigned carry;
  if (threadIdx.x == 0) carry = 0;
  __syncthreads();
  for (int base = 0; base < N_NODES; base += 1024) {
    const int i = base + threadIdx.x;
    const unsigned v = (i < N_NODES) ? cnt[i] : 0u;
    buf[threadIdx.x] = v;
    __syncthreads();
    for (int off = 1; off < 1024; off <<= 1) {      // Hillis-Steele inclusive
      unsigned t = (threadIdx.x >= (unsigned)off) ? buf[threadIdx.x - off] : 0u;
      __syncthreads();
      buf[threadIdx.x] += t;
      __syncthreads();
    }
    if (i < N_NODES) rowstart[i] = carry + buf[threadIdx.x] - v;  // exclusive
    __syncthreads();
    if (threadIdx.x == 0) carry += buf[1023];
    __syncthreads();
  }
  if (threadIdx.x == 0) rowstart[N_NODES] = carry;
}

__global__ void k_place(const int* __restrict__ dst,
                        const unsigned* __restrict__ rowstart,
                        unsigned* __restrict__ cursor,
                        unsigned* __restrict__ csrpos) {
  int e = blockIdx.x * blockDim.x + threadIdx.x;
  if (e < N_EDGES) {
    const int d = dst[e];
    const unsigned p = atomicAdd(&cursor[d], 1u);
    csrpos[e] = rowstart[d] + p;
  }
}

// ==== fold edge encoder into each layer's pretrans weight ====================
__global__ __launch_bounds__(256) void k_fuse(const float* __restrict__ w_edge,
                                              const float* __restrict__ b_edge,
                                              const float* __restrict__ w_pre,
                                              const float* __restrict__ b_pre,
                                              float* __restrict__ Wf,
                                              float* __restrict__ bf) {
  const int l = blockIdx.x;
  const float* wp = w_pre + (size_t)l * KPRE * 64;
  float* wf = Wf + (size_t)l * KPRE * 64;
  const int t = threadIdx.x;
  for (int i = t; i < 128 * 64; i += 256) wf[i] = wp[i];
  for (int i = t; i < 64 * 64; i += 256) {
    const int r = i >> 6, c = i & 63;
    float s = 0.f;
    for (int j = 0; j < 64; ++j) s += w_edge[r * 64 + j] * wp[(128 + j) * 64 + c];
    wf[(128 + r) * 64 + c] = s;
  }
  if (t < 64) {
    float s = b_pre[l * 64 + t];
    for (int j = 0; j < 64; ++j) s += b_edge[j] * wp[(128 + j) * 64 + t];
    bf[l * 64 + t] = s;
  }
}

// ---- node encoder: [M x 64] @ [64 x 64] + bias ----
__global__ __launch_bounds__(128) void k_encode64(const float* __restrict__ X,
                                                  const float2* __restrict__ Wp,
                                                  const float* __restrict__ b,
                                                  float* __restrict__ Y) {
  __shared__ float at[16 * 68];
  const int mt = blockIdx.x * 16;
  const int t  = threadIdx.x;
  for (int i = t; i < 16 * 16; i += 128) {
    int row = i >> 4, j = i & 15;
    ((float4*)(at + row * 68))[j] = ((const float4*)(X + (size_t)(mt + row) * 64))[j];
  }
  __syncthreads();
  const int wave = t >> 5;
  v8f c = wmma_gemm_pk(at, 68, Wp, 64, wave * 16, 64);
  const int lane = t & 31;
  const int col  = wave * 16 + (lane & 15);
  const int rb   = (lane >> 4) * 8;
#pragma unroll
  for (int i = 0; i < 8; ++i)
    Y[(size_t)(mt + rb + i) * 64 + col] = c[i] + b[col];
}

// ---- pretrans: persistent, double-buffered async-to-LDS gather + WMMA -------
// z row = [h[src] | h[dst] | edge_feat], K=192; output row scattered to
// dst-sorted slot csrpos[e].
__global__ __launch_bounds__(128) void k_pretrans(
    const float* __restrict__ h, const float* __restrict__ efeat,
    const int* __restrict__ src, const int* __restrict__ dst,
    const unsigned* __restrict__ csrpos,
    const float2* __restrict__ Wp, const float* __restrict__ bp,
    float* __restrict__ mout) {
  __shared__ float zt[2][16 * 196];            // double-buffered 16x(192+pad)
  const int t = threadIdx.x;
  const int tile0 = blockIdx.x * TPB_TILES;

  // async gather of one 16-edge tile into buffer b (ASYNCcnt-tracked DMA)
  auto issue = [&](int tile, int b) {
    const int et = tile * 16;
#pragma unroll
    for (int i = 0; i < 6; ++i) {
      const int j  = t + 128 * i;              // 768 float4 slots
      const int r  = j / 48, c4 = j % 48;
      const int e  = et + r;
      const float* g;
      if (c4 < 16)      g = h + (size_t)src[e] * 64 + c4 * 4;
      else if (c4 < 32) g = h + (size_t)dst[e] * 64 + (c4 - 16) * 4;
      else              g = efeat + (size_t)e * 64 + (c4 - 32) * 4;
      const unsigned lo = (unsigned)(uintptr_t)&zt[b][r * 196 + c4 * 4];
      asm volatile("global_load_async_to_lds_b128 %0, %1, off"
                   :: "v"(lo), "v"(g) : "memory");
    }
  };

  issue(tile0, 0);
  for (int k = 0; k < TPB_TILES; ++k) {
    const int cur = k & 1;
    asm volatile("s_wait_asynccnt 0" ::: "memory");   // our tile data landed
    __syncthreads();                                   // all waves' parts landed
    if (k + 1 < TPB_TILES) issue(tile0 + k + 1, cur ^ 1);  // overlap next gather

    const int wave = t >> 5;
    v8f c = wmma_gemm_pk(zt[cur], 196, Wp, 64, wave * 16, KPRE);
    const int lane = t & 31;
    const int col  = wave * 16 + (lane & 15);
    const int rb   = (lane >> 4) * 8;
    const float bb = bp[col];
    const int et   = (tile0 + k) * 16;
#pragma unroll
    for (int i = 0; i < 8; ++i) {
      const unsigned pos = csrpos[et + rb + i];
      mout[(size_t)pos * 64 + col] = c[i] + bb;
    }
    __syncthreads();                 // buffer free before re-issue next iter
  }
}

// ---- posttrans: CSR segment-reduce messages, build 13H row, WMMA, residual ----
__global__ __launch_bounds__(128) void k_posttrans(
    const float* __restrict__ h, const unsigned* __restrict__ rowstart,
    const float* __restrict__ m,
    const float2* __restrict__ Wq, const float* __restrict__ bq,
    float* __restrict__ hout) {
  __shared__ float zt[16 * 836];
  const int nt = blockIdx.x * 16;
  const int t  = threadIdx.x;
  const int r  = t >> 3, q = t & 7;            // 8 threads per node, 8 cols each
  const int node = nt + r;
  const unsigned e0 = rowstart[node], e1 = rowstart[node + 1];
  const float dg = (float)(e1 - e0);
  const bool has = e1 > e0;

  float s1[8], s2[8], mx[8], mn[8];
#pragma unroll
  for (int i = 0; i < 8; ++i) {
    s1[i] = 0.f; s2[i] = 0.f; mx[i] = -INFINITY; mn[i] = INFINITY;
  }
  for (unsigned j = e0; j < e1; ++j) {         // contiguous message rows
    const float4* mr = (const float4*)(m + (size_t)j * 64 + q * 8);
    const float4 a = mr[0], b = mr[1];
    const float v[8] = {a.x, a.y, a.z, a.w, b.x, b.y, b.z, b.w};
#pragma unroll
    for (int i = 0; i < 8; ++i) {
      const float x = v[i];
      s1[i] += x; s2[i] += x * x;
      mx[i] = fmaxf(mx[i], x); mn[i] = fminf(mn[i], x);
    }
  }
  const float dsafe = fmaxf(dg, 1.f);
  const float ld    = logf(dg + 1.f);
  const float sc2   = ld * (1.f / AVG_D_LOG);
  const float sc3   = AVG_D_LOG / (has ? ld : 1.f);
  float* zr = zt + r * 836;
#pragma unroll
  for (int i = 0; i < 8; ++i) {
    const int cc = q * 8 + i;
    const float mean = s1[i] / dsafe;
    const float var  = s2[i] / dsafe - mean * mean;
    const float sd   = sqrtf(fmaxf(var, 0.f) + EPSV);
    const float m0 = has ? mean : 0.f;
    const float m1 = has ? mx[i] : 0.f;
    const float m2 = has ? mn[i] : 0.f;
    const float m3 = has ? sd : 0.f;
    zr[cc]        = h[(size_t)node * 64 + cc];
    zr[64 + cc]   = m0;        zr[128 + cc] = m1;
    zr[192 + cc]  = m2;        zr[256 + cc] = m3;
    zr[320 + cc]  = m0 * sc2;  zr[384 + cc] = m1 * sc2;
    zr[448 + cc]  = m2 * sc2;  zr[512 + cc] = m3 * sc2;
    zr[576 + cc]  = m0 * sc3;  zr[640 + cc] = m1 * sc3;
    zr[704 + cc]  = m2 * sc3;  zr[768 + cc] = m3 * sc3;
  }
  __syncthreads();
  const int wave = t >> 5;
  v8f c = wmma_gemm_pk(zt, 836, Wq, 64, wave * 16, KPOST);
  const int lane = t & 31;
  const int col  = wave * 16 + (lane & 15);
  const int rb   = (lane >> 4) * 8;
#pragma unroll
  for (int i = 0; i < 8; ++i) {
    const int nd = nt + rb + i;
    hout[(size_t)nd * 64 + col] = c[i] + bq[col] + h[(size_t)nd * 64 + col];
  }
}

// ---- fused readout: relu(relu(h@W1+b1)@W2+b2) -> [N, 256] ----
__global__ __launch_bounds__(128) void k_readout(const float* __restrict__ h,
                                                 const float2* __restrict__ W1p,
                                                 const float* __restrict__ b1,
                                                 const float2* __restrict__ W2p,
                                                 const float* __restrict__ b2,
                                                 float* __restrict__ out) {
  __shared__ float at[16 * 68];
  __shared__ float tt[16 * 68];
  const int mt = blockIdx.x * 16;
  const int t  = threadIdx.x;
  for (int i = t; i < 16 * 16; i += 128) {
    int row = i >> 4, j = i & 15;
    ((float4*)(at + row * 68))[j] = ((const float4*)(h + (size_t)(mt + row) * 64))[j];
  }
  __syncthreads();
  const int wave = t >> 5;
  const int lane = t & 31;
  const int rb   = (lane >> 4) * 8;
  {
    v8f c = wmma_gemm_pk(at, 68, W1p, 64, wave * 16, 64);
    const int col = wave * 16 + (lane & 15);
#pragma unroll
    for (int i = 0; i < 8; ++i)
      tt[(rb + i) * 68 + col] = fmaxf(c[i] + b1[col], 0.f);
  }
  __syncthreads();
#pragma unroll
  for (int j = 0; j < 4; ++j) {                // 4 waves x 4 tiles = 256 cols
    const int ct = wave * 4 + j;
    v8f c = wmma_gemm_pk(tt, 68, W2p, 256, ct * 16, 64);
    const int col = ct * 16 + (lane & 15);
#pragma unroll
    for (int i = 0; i < 8; ++i)
      out[(size_t)(mt + rb + i) * 256 + col] = fmaxf(c[i] + b2[col], 0.f);
  }
}

extern "C" void kernel_launch(void* const* d_in, const int* in_sizes, int n_in,
                              void* d_out, int out_size, void* d_ws, size_t ws_size,
                              hipStream_t stream) {
  const float* node_feat = (const float*)d_in[0];
  const float* edge_feat = (const float*)d_in[1];
  const int*   src       = (const int*)d_in[2];
  const int*   dst       = (const int*)d_in[3];
  const float* w_node    = (const float*)d_in[4];
  const float* b_node    = (const float*)d_in[5];
  const float* w_edge    = (const float*)d_in[6];
  const float* b_edge    = (const float*)d_in[7];
  const float* w_pre     = (const float*)d_in[8];
  const float* b_pre     = (const float*)d_in[9];
  const float* w_post    = (const float*)d_in[10];
  const float* b_post    = (const float*)d_in[11];
  const float* w_r1      = (const float*)d_in[12];
  const float* b_r1      = (const float*)d_in[13];
  const float* w_r2      = (const float*)d_in[14];
  const float* b_r2      = (const float*)d_in[15];

  const size_t N64 = (size_t)N_NODES * 64;
  const size_t E64 = (size_t)N_EDGES * 64;
  float* h0 = (float*)d_ws;
  float* h1 = h0 + N64;
  float* m  = h1 + N64;                        // dst-sorted messages [E,64]
  float* Wf = m + E64;                         // fused pretrans weights [L,192,64]
  float* bf = Wf + (size_t)NLAYERS * KPRE * 64;
  float2* pWnode = (float2*)(bf + NLAYERS * 64);           // 2048 f2
  float2* pW1    = pWnode + 2048;                          // 2048 f2
  float2* pW2    = pW1 + 2048;                             // 8192 f2
  float2* pWf    = pW2 + 8192;                             // 5*6144 f2
  float2* pWpost = pWf + (size_t)NLAYERS * 6144;           // 5*26624 f2
  unsigned* cnt      = (unsigned*)(pWpost + (size_t)NLAYERS * 26624);
  unsigned* rowstart = cnt + N_NODES;          // [N+1]
  unsigned* cursor   = rowstart + (N_NODES + 1);
  unsigned* csrpos   = cursor + N_NODES;       // [E]

  // --- CSR build (once per call) ---
  hipMemsetAsync(cnt, 0, N_NODES * sizeof(unsigned), stream);
  hipMemsetAsync(cursor, 0, N_NODES * sizeof(unsigned), stream);
  k_hist<<<N_EDGES / 256, 256, 0, stream>>>(dst, cnt);
  k_scan<<<1, 1024, 0, stream>>>(cnt, rowstart);
  k_place<<<N_EDGES / 256, 256, 0, stream>>>(dst, rowstart, cursor, csrpos);

  // --- fold edge encoder into per-layer pretrans weights, then pack all B ---
  k_fuse<<<NLAYERS, 256, 0, stream>>>(w_edge, b_edge, w_pre, b_pre, Wf, bf);
  k_pack<<<8,  256, 0, stream>>>(w_node, pWnode, 64, 64);
  k_pack<<<8,  256, 0, stream>>>(w_r1, pW1, 64, 64);
  k_pack<<<32, 256, 0, stream>>>(w_r2, pW2, 64, 256);
  for (int l = 0; l < NLAYERS; ++l) {
    k_pack<<<24,  256, 0, stream>>>(Wf + (size_t)l * KPRE * 64,
                                    pWf + (size_t)l * 6144, KPRE, 64);
    k_pack<<<104, 256, 0, stream>>>(w_post + (size_t)l * KPOST * 64,
                                    pWpost + (size_t)l * 26624, KPOST, 64);
  }

  // --- node encoder ---
  k_encode64<<<N_NODES / 16, 128, 0, stream>>>(node_feat, pWnode, b_node, h0);

  float* hin  = h0;
  float* hout = h1;
  for (int l = 0; l < NLAYERS; ++l) {
    k_pretrans<<<N_EDGES / (16 * TPB_TILES), 128, 0, stream>>>(
        hin, edge_feat, src, dst, csrpos, pWf + (size_t)l * 6144,
        bf + (size_t)l * 64, m);
    k_posttrans<<<N_NODES / 16, 128, 0, stream>>>(
        hin, rowstart, m, pWpost + (size_t)l * 26624,
        b_post + (size_t)l * 64, hout);
    float* tmp = hin; hin = hout; hout = tmp;
  }

  k_readout<<<N_NODES / 16, 128, 0, stream>>>(hin, pW1, b_r1, pW2, b_r2,
                                              (float*)d_out);
}